// GNNCF_37898791420448
// MI455X (gfx1250) — compile-verified
//
#include <hip/hip_runtime.h>
#include <hip/hip_bf16.h>
#include <math.h>

// ---------------------------------------------------------------------------
// NGCF GNN-CF forward for MI455X (gfx1250, wave32).
//   N = 100000 nodes, EMB = 64, E = 3.2M edges, 2 layers, out = [N, 192] f32.
// Strategy: SpMM via hardware f32 atomics (node table is L2-resident: 25.6MB
// of 192MB L2); fused dual-GEMM [N,128]@[128,64] via V_WMMA_F32_16X16X4_F32
// to keep full fp32 precision.
// ---------------------------------------------------------------------------

typedef __attribute__((ext_vector_type(2))) float v2f;
typedef __attribute__((ext_vector_type(8))) float v8f;

#define GNN_N       100000
#define GNN_NUSERS  60000
#define GNN_EMB     64
#define GNN_E       3200000

// ---------------------------------------------------------------------------
// ego = concat(user_emb, item_emb); also write layer-0 slice of output.
// ---------------------------------------------------------------------------
__global__ __launch_bounds__(256) void gnn_init(
    const float* __restrict__ user_emb, const float* __restrict__ item_emb,
    float* __restrict__ ego, float* __restrict__ out)
{
    long tid = (long)blockIdx.x * 256 + threadIdx.x;
    if (tid >= (long)GNN_N * GNN_EMB) return;
    long row = tid >> 6;
    int  col = (int)(tid & 63);
    float v = (row < GNN_NUSERS)
                ? user_emb[row * GNN_EMB + col]
                : item_emb[(row - GNN_NUSERS) * GNN_EMB + col];
    ego[tid] = v;
    out[row * 192 + col] = v;   // all_embs[0]
}

__global__ __launch_bounds__(256) void gnn_zero(float4* __restrict__ p, long n4)
{
    long tid = (long)blockIdx.x * 256 + threadIdx.x;
    if (tid < n4) p[tid] = make_float4(0.f, 0.f, 0.f, 0.f);
}

// ---------------------------------------------------------------------------
// SpMM: y[src[e]] += val[e] * x[dst[e]]   (row = 64 f32)
// 8 threads per edge, 8 floats each -> hardware global_atomic_add_f32.
// ---------------------------------------------------------------------------
__global__ __launch_bounds__(256) void gnn_spmm(
    const int* __restrict__ esrc, const int* __restrict__ edst,
    const float* __restrict__ eval, const float* __restrict__ x,
    float* __restrict__ y)
{
    long tid = (long)blockIdx.x * 256 + threadIdx.x;
    long e = tid >> 3;
    if (e >= GNN_E) return;
    int part = (int)(tid & 7);
    int d = edst[e];
    int s = esrc[e];
    float v = eval[e];
    const float4* xr = (const float4*)(x + (long)d * GNN_EMB) + part * 2;
    float4 a = xr[0];
    float4 b = xr[1];
    float* yr = y + (long)s * GNN_EMB + part * 8;
    unsafeAtomicAdd(yr + 0, v * a.x);
    unsafeAtomicAdd(yr + 1, v * a.y);
    unsafeAtomicAdd(yr + 2, v * a.z);
    unsafeAtomicAdd(yr + 3, v * a.w);
    unsafeAtomicAdd(yr + 4, v * b.x);
    unsafeAtomicAdd(yr + 5, v * b.y);
    unsafeAtomicAdd(yr + 6, v * b.z);
    unsafeAtomicAdd(yr + 7, v * b.w);
}

// ---------------------------------------------------------------------------
// Fused layer:
//   side_A = side_L + ego ; prod = ego * side_L
//   act    = leaky_relu( [side_A | prod] @ [W_gc ; W_bi] + (b_gc + b_bi) )
//   ego   <- act (in place) ; out[:, col0:col0+64] = act / max(||act||, 1e-12)
//
// 8 waves / block, one 16-row tile per wave, 4 N-tiles of 16, K = 128 via
// 32 steps of V_WMMA_F32_16X16X4_F32 (A: 16x4 f32 -> v2f per lane;
// lanes 0-15: M=lane, K=4s+{0,1}; lanes 16-31: M=lane-16, K=4s+{2,3}).
// ---------------------------------------------------------------------------
__global__ __launch_bounds__(256) void gnn_layer(
    const float* __restrict__ sideL, float* __restrict__ ego,
    const float* __restrict__ Wgc, const float* __restrict__ bgc,
    const float* __restrict__ Wbi, const float* __restrict__ bbi,
    float* __restrict__ out, int out_col0)
{
    __shared__ float Wt[64][132];   // Wt[n][k]: k<64 -> W_gc[k][n], k>=64 -> W_bi[k-64][n]
    __shared__ float bsum[64];

    const int tid = threadIdx.x;

    // Stage combined weights transposed into LDS (coalesced global reads).
    for (int i = tid; i < 64 * 128; i += 256) {
        int k = i >> 6;
        int n = i & 63;
        float w = (k < 64) ? Wgc[k * 64 + n] : Wbi[(k - 64) * 64 + n];
        Wt[n][k] = w;
    }
    if (tid < 64) bsum[tid] = bgc[tid] + bbi[tid];
    __syncthreads();

    const int wave    = tid >> 5;
    const int lane    = tid & 31;
    const int rowbase = blockIdx.x * 128 + wave * 16;
    if (rowbase >= GNN_N) return;          // uniform per wave; EXEC stays full

    const int m    = lane & 15;            // A-row / B-col index within tile
    const int half = lane >> 4;            // 0 or 1
    const int kb   = half << 1;            // K sub-offset: 0 or 2

    const long  row    = (long)(rowbase + m);
    const float* egoRd = ego   + row * 64;
    const float* sLRd  = sideL + row * 64;

    // Preload exactly the K-slices this lane's A fragments need.
    v2f e2[16], l2[16];
#pragma unroll
    for (int s = 0; s < 16; ++s) {
        int c = 4 * s + kb;
        e2[s] = *(const v2f*)(egoRd + c);
        l2[s] = *(const v2f*)(sLRd + c);
    }

    v8f acc[4] = {};

    // K = 0..63 : side_A = side_L + ego
#pragma unroll
    for (int s = 0; s < 16; ++s) {
        v2f a;
        a.x = e2[s].x + l2[s].x;
        a.y = e2[s].y + l2[s].y;
#pragma unroll
        for (int nt = 0; nt < 4; ++nt) {
            v2f b = *(const v2f*)(&Wt[nt * 16 + m][4 * s + kb]);
            acc[nt] = __builtin_amdgcn_wmma_f32_16x16x4_f32(
                false, a, false, b, (short)0, acc[nt], false, false);
        }
    }
    // K = 64..127 : prod = ego * side_L
#pragma unroll
    for (int s = 0; s < 16; ++s) {
        v2f a;
        a.x = e2[s].x * l2[s].x;
        a.y = e2[s].y * l2[s].y;
#pragma unroll
        for (int nt = 0; nt < 4; ++nt) {
            v2f b = *(const v2f*)(&Wt[nt * 16 + m][64 + 4 * s + kb]);
            acc[nt] = __builtin_amdgcn_wmma_f32_16x16x4_f32(
                false, a, false, b, (short)0, acc[nt], false, false);
        }
    }

    // Epilogue: bias + leaky_relu(0.01); D layout: VGPR j -> row (j + 8*half),
    // col (nt*16 + m).
    float vout[4][8];
    float ssq[8];
#pragma unroll
    for (int j = 0; j < 8; ++j) ssq[j] = 0.f;
#pragma unroll
    for (int nt = 0; nt < 4; ++nt) {
        float bc = bsum[nt * 16 + m];
#pragma unroll
        for (int j = 0; j < 8; ++j) {
            float v = acc[nt][j] + bc;
            v = (v > 0.f) ? v : 0.01f * v;
            vout[nt][j] = v;
            ssq[j] += v * v;
        }
    }

    // Write next-layer ego in place (each row owned by exactly one wave).
    float* egoWr = ego + (long)rowbase * 64;
#pragma unroll
    for (int j = 0; j < 8; ++j) {
        int mrow = j + (half << 3);
#pragma unroll
        for (int nt = 0; nt < 4; ++nt)
            egoWr[mrow * 64 + nt * 16 + m] = vout[nt][j];
    }

    // Row L2 norms: reduce across the 16 lanes holding each row (wave32-safe).
#pragma unroll
    for (int j = 0; j < 8; ++j) {
        float s = ssq[j];
        s += __shfl_xor(s, 8, 16);
        s += __shfl_xor(s, 4, 16);
        s += __shfl_xor(s, 2, 16);
        s += __shfl_xor(s, 1, 16);
        float inv = 1.0f / fmaxf(sqrtf(s), 1e-12f);
        int mrow = j + (half << 3);
        float* o = out + (long)(rowbase + mrow) * 192 + out_col0;
#pragma unroll
        for (int nt = 0; nt < 4; ++nt)
            o[nt * 16 + m] = vout[nt][j] * inv;
    }
}

// ---------------------------------------------------------------------------
extern "C" void kernel_launch(void* const* d_in, const int* in_sizes, int n_in,
                              void* d_out, int out_size, void* d_ws, size_t ws_size,
                              hipStream_t stream) {
    const float* user_emb = (const float*)d_in[0];
    const float* item_emb = (const float*)d_in[1];
    const float* W_gc0 = (const float*)d_in[2];
    const float* b_gc0 = (const float*)d_in[3];
    const float* W_bi0 = (const float*)d_in[4];
    const float* b_bi0 = (const float*)d_in[5];
    const float* W_gc1 = (const float*)d_in[6];
    const float* b_gc1 = (const float*)d_in[7];
    const float* W_bi1 = (const float*)d_in[8];
    const float* b_bi1 = (const float*)d_in[9];
    const int*   e_src = (const int*)d_in[10];
    const int*   e_dst = (const int*)d_in[11];
    const float* e_val = (const float*)d_in[12];
    float* out = (float*)d_out;

    float* ego   = (float*)d_ws;                        // [N,64]
    float* sideL = ego + (long)GNN_N * GNN_EMB;         // [N,64]

    const long nelem  = (long)GNN_N * GNN_EMB;          // 6.4M
    const int  gInit  = (int)((nelem + 255) / 256);
    const int  gZero  = (int)((nelem / 4 + 255) / 256);
    const int  gSpmm  = (int)(((long)GNN_E * 8 + 255) / 256);
    const int  gLayer = (GNN_N + 127) / 128;

    gnn_init<<<gInit, 256, 0, stream>>>(user_emb, item_emb, ego, out);

    // ---- layer 0 ----
    gnn_zero<<<gZero, 256, 0, stream>>>((float4*)sideL, nelem / 4);
    gnn_spmm<<<gSpmm, 256, 0, stream>>>(e_src, e_dst, e_val, ego, sideL);
    gnn_layer<<<gLayer, 256, 0, stream>>>(sideL, ego, W_gc0, b_gc0, W_bi0, b_bi0,
                                          out, 64);

    // ---- layer 1 ----
    gnn_zero<<<gZero, 256, 0, stream>>>((float4*)sideL, nelem / 4);
    gnn_spmm<<<gSpmm, 256, 0, stream>>>(e_src, e_dst, e_val, ego, sideL);
    gnn_layer<<<gLayer, 256, 0, stream>>>(sideL, ego, W_gc1, b_gc1, W_bi1, b_bi1,
                                          out, 128);
}